// LogitGenerator_21191368638517
// MI455X (gfx1250) — compile-verified
//
#include <hip/hip_runtime.h>
#include <hip/hip_bf16.h>
#include <math.h>

typedef __attribute__((ext_vector_type(2))) float v2f;
typedef __attribute__((ext_vector_type(4))) float v4f;
typedef __attribute__((ext_vector_type(8))) float v8f;

#define DE     768     // encoder dim (K of projection GEMM)
#define DF     256     // projected dim (K of logits GEMM)
#define NC     500     // classes
#define NCP    512     // padded class count
#define NROWS  24000   // gathered rows (12000 per mask, concatenated)
#define OUTW   501     // C + 1
#define INV_TEMP 10.0f // 1 / 0.1
#define NEPS   1e-8f

#define KB1    (DE / 4)   // 192 k-blocks, projection GEMM
#define KB2    (DF / 4)   // 64 k-blocks, logits GEMM
#define LDSTR  68         // padded LDS row stride (floats)

__device__ __forceinline__ float wave_sum32(float v) {
    #pragma unroll
    for (int m = 16; m > 0; m >>= 1) v += __shfl_xor(v, m, 32);
    return v;
}

// ---------------- Kernel 0a: normalize emb rows -> embn
__global__ __launch_bounds__(32) void emb_norm_kernel(const float* __restrict__ emb,
                                                      float* __restrict__ embn) {
    int c = blockIdx.x;          // 0..499
    int lane = threadIdx.x;      // 0..31
    float vals[8];
    float s = 0.f;
    #pragma unroll
    for (int i = 0; i < 8; i++) {
        float v = emb[(size_t)c * DF + lane + 32 * i];
        vals[i] = v;
        s += v * v;
    }
    s = wave_sum32(s);
    float inv = 1.f / fmaxf(sqrtf(s), NEPS);
    #pragma unroll
    for (int i = 0; i < 8; i++) embn[(size_t)c * DF + lane + 32 * i] = vals[i] * inv;
}

// ---------------- Kernel 0b: exact row-equality table (reference's neg_is_pos)
__global__ __launch_bounds__(256) void row_eq_kernel(const float* __restrict__ emb,
                                                     unsigned char* __restrict__ row_eq) {
    int id = blockIdx.x * 256 + threadIdx.x;
    if (id >= NC * NC) return;
    int c1 = id / NC, c2 = id % NC;
    bool eq = true;
    for (int d = 0; d < DF; d++) {
        if (emb[(size_t)c1 * DF + d] != emb[(size_t)c2 * DF + d]) { eq = false; break; }
    }
    row_eq[id] = eq ? 1u : 0u;
}

// ---------------- Kernel 0c: pack W into B-fragment order
// Wpack[kb][nb(4 waves)][lane(32)][t(4)][j(2)] = W[(kb*4 + 2*(lane>>4) + j)][nb*64 + t*16 + (lane&15)]
__global__ __launch_bounds__(256) void pack_w_kernel(const float* __restrict__ W,
                                                     float* __restrict__ Wpack) {
    int id = blockIdx.x * 256 + threadIdx.x;       // 192*4*32*8 = 196608
    int j    = id & 1;
    int t    = (id >> 1) & 3;
    int lane = (id >> 3) & 31;
    int nb   = (id >> 8) & 3;
    int kb   = id >> 10;
    int k = kb * 4 + 2 * (lane >> 4) + j;
    int n = nb * 64 + t * 16 + (lane & 15);
    Wpack[id] = W[(size_t)k * DF + n];
}

// ---------------- Kernel 0d: pack embn^T (padded to 512 cols) into B-fragment order
// Epack[kb][nb(4 waves)][lane(32)][t(8)][j(2)] = embnT[k][n] = embn[n][k], 0 if n>=500
__global__ __launch_bounds__(256) void pack_e_kernel(const float* __restrict__ embn,
                                                     float* __restrict__ Epack) {
    int id = blockIdx.x * 256 + threadIdx.x;       // 64*4*32*16 = 131072
    int j    = id & 1;
    int t    = (id >> 1) & 7;
    int lane = (id >> 4) & 31;
    int nb   = (id >> 9) & 3;
    int kb   = id >> 11;
    int k = kb * 4 + 2 * (lane >> 4) + j;
    int n = nb * 128 + t * 16 + (lane & 15);
    Epack[id] = (n < NC) ? embn[(size_t)n * DF + k] : 0.f;
}

// ---------------- Kernel 1: gathered projection GEMM  pg[r] = x[mask[r]] @ W + b
// grid = 1500 M-tiles of 16 rows; 4 waves/block, each wave owns 64 cols (4 N-tiles).
// A tile staged through LDS (16 rows x 64 K per chunk), B streamed packed.
__global__ __launch_bounds__(128) void proj_gemm_kernel(const float* __restrict__ x,
                                                        const int* __restrict__ mask_m,
                                                        const int* __restrict__ mask_u,
                                                        const float* __restrict__ Wpack,
                                                        const float* __restrict__ bias,
                                                        float* __restrict__ pg) {
    __shared__ float As[16 * LDSTR];
    int tile_m = blockIdx.x;
    int tid  = threadIdx.x;
    int wave = tid >> 5;
    int lane = tid & 31;
    int lm   = lane & 15;
    int half = lane >> 4;

    // cooperative A-load assignment: 8 threads per row, 8 floats per thread
    int ldrow = tid >> 3;
    int ldcol = (tid & 7) * 8;
    int grow  = tile_m * 16 + ldrow;
    int src   = (grow < 12000) ? mask_m[grow] : mask_u[grow - 12000];
    const float* xrow = x + (size_t)src * DE;

    const float* wp = Wpack + ((size_t)wave * 32 + lane) * 8;   // + kb*1024
    v8f acc[4] = {};

    for (int c = 0; c < DE / 64; c++) {           // 12 chunks of 64 K
        v4f f0 = *(const v4f*)(xrow + c * 64 + ldcol);
        v4f f1 = *(const v4f*)(xrow + c * 64 + ldcol + 4);
        __syncthreads();                          // prior chunk's LDS reads done
        *(v4f*)(&As[ldrow * LDSTR + ldcol])     = f0;
        *(v4f*)(&As[ldrow * LDSTR + ldcol + 4]) = f1;
        __syncthreads();                          // LDS chunk visible
        #pragma unroll
        for (int kk = 0; kk < 64; kk += 4) {
            int kb = c * 16 + (kk >> 2);
            v2f a = *(const v2f*)(&As[lm * LDSTR + kk + 2 * half]);
            const float* bp = wp + (size_t)kb * 1024;
            v4f q0 = *(const v4f*)(bp);
            v4f q1 = *(const v4f*)(bp + 4);
            v2f b0{q0.x, q0.y}, b1{q0.z, q0.w}, b2{q1.x, q1.y}, b3{q1.z, q1.w};
            acc[0] = __builtin_amdgcn_wmma_f32_16x16x4_f32(false, a, false, b0, (short)0, acc[0], false, false);
            acc[1] = __builtin_amdgcn_wmma_f32_16x16x4_f32(false, a, false, b1, (short)0, acc[1], false, false);
            acc[2] = __builtin_amdgcn_wmma_f32_16x16x4_f32(false, a, false, b2, (short)0, acc[2], false, false);
            acc[3] = __builtin_amdgcn_wmma_f32_16x16x4_f32(false, a, false, b3, (short)0, acc[3], false, false);
        }
    }

    #pragma unroll
    for (int t = 0; t < 4; t++) {
        int n = wave * 64 + t * 16 + lm;
        float bv = bias[n];
        #pragma unroll
        for (int v = 0; v < 8; v++) {
            int r = tile_m * 16 + v + 8 * half;
            pg[(size_t)r * DF + n] = acc[t][v] + bv;
        }
    }
}

// ---------------- Kernel 2: per-row inverse norm, target id, and pos logit (col 0)
__global__ __launch_bounds__(256) void norm_pos_kernel(const float* __restrict__ pg,
                                                       const int* __restrict__ label,
                                                       const int* __restrict__ mask_m,
                                                       const int* __restrict__ mask_u,
                                                       const float* __restrict__ embn,
                                                       float* __restrict__ rscale,
                                                       int* __restrict__ tgtw,
                                                       float* __restrict__ out) {
    int f = blockIdx.x * 8 + (threadIdx.x >> 5);
    int lane = threadIdx.x & 31;
    const float* prow = pg + (size_t)f * DF + lane * 8;

    v4f p0 = *(const v4f*)(prow);
    v4f p1 = *(const v4f*)(prow + 4);
    float s = p0.x*p0.x + p0.y*p0.y + p0.z*p0.z + p0.w*p0.w
            + p1.x*p1.x + p1.y*p1.y + p1.z*p1.z + p1.w*p1.w;
    s = wave_sum32(s);

    int midx = (f < 12000) ? mask_m[f] : mask_u[f - 12000];
    int t = label[midx];
    const float* erow = embn + (size_t)t * DF + lane * 8;
    v4f e0 = *(const v4f*)(erow);
    v4f e1 = *(const v4f*)(erow + 4);
    float d = p0.x*e0.x + p0.y*e0.y + p0.z*e0.z + p0.w*e0.w
            + p1.x*e1.x + p1.y*e1.y + p1.z*e1.z + p1.w*e1.w;
    d = wave_sum32(d);

    float inv = 1.f / fmaxf(sqrtf(s), NEPS);
    if (lane == 0) {
        rscale[f] = inv;
        tgtw[f]   = t;
        out[(size_t)f * OUTW] = d * inv * INV_TEMP;   // pos / temp
    }
}

// ---------------- Kernel 3: logits GEMM  raw = pg @ embn^T, scaled + masked epilogue
// grid = 1500 M-tiles; 4 waves/block, each wave owns 128 cols (8 N-tiles over 512 padded)
__global__ __launch_bounds__(128) void logit_gemm_kernel(const float* __restrict__ pg,
                                                         const float* __restrict__ Epack,
                                                         const float* __restrict__ rscale,
                                                         const int* __restrict__ tgtw,
                                                         const unsigned char* __restrict__ row_eq,
                                                         float* __restrict__ out) {
    __shared__ float As[16 * LDSTR];
    int tile_m = blockIdx.x;
    int tid  = threadIdx.x;
    int wave = tid >> 5;
    int lane = tid & 31;
    int lm   = lane & 15;
    int half = lane >> 4;

    int ldrow = tid >> 3;
    int ldcol = (tid & 7) * 8;
    const float* arow = pg + (size_t)(tile_m * 16 + ldrow) * DF;

    const float* ep = Epack + ((size_t)wave * 32 + lane) * 16;   // + kb*4096
    v8f acc[8] = {};

    for (int c = 0; c < DF / 64; c++) {           // 4 chunks of 64 K
        v4f f0 = *(const v4f*)(arow + c * 64 + ldcol);
        v4f f1 = *(const v4f*)(arow + c * 64 + ldcol + 4);
        __syncthreads();
        *(v4f*)(&As[ldrow * LDSTR + ldcol])     = f0;
        *(v4f*)(&As[ldrow * LDSTR + ldcol + 4]) = f1;
        __syncthreads();
        #pragma unroll
        for (int kk = 0; kk < 64; kk += 4) {
            int kb = c * 16 + (kk >> 2);
            v2f a = *(const v2f*)(&As[lm * LDSTR + kk + 2 * half]);
            const float* bp = ep + (size_t)kb * 4096;
            v4f q0 = *(const v4f*)(bp);
            v4f q1 = *(const v4f*)(bp + 4);
            v4f q2 = *(const v4f*)(bp + 8);
            v4f q3 = *(const v4f*)(bp + 12);
            v2f b0{q0.x, q0.y}, b1{q0.z, q0.w}, b2{q1.x, q1.y}, b3{q1.z, q1.w};
            v2f b4{q2.x, q2.y}, b5{q2.z, q2.w}, b6{q3.x, q3.y}, b7{q3.z, q3.w};
            acc[0] = __builtin_amdgcn_wmma_f32_16x16x4_f32(false, a, false, b0, (short)0, acc[0], false, false);
            acc[1] = __builtin_amdgcn_wmma_f32_16x16x4_f32(false, a, false, b1, (short)0, acc[1], false, false);
            acc[2] = __builtin_amdgcn_wmma_f32_16x16x4_f32(false, a, false, b2, (short)0, acc[2], false, false);
            acc[3] = __builtin_amdgcn_wmma_f32_16x16x4_f32(false, a, false, b3, (short)0, acc[3], false, false);
            acc[4] = __builtin_amdgcn_wmma_f32_16x16x4_f32(false, a, false, b4, (short)0, acc[4], false, false);
            acc[5] = __builtin_amdgcn_wmma_f32_16x16x4_f32(false, a, false, b5, (short)0, acc[5], false, false);
            acc[6] = __builtin_amdgcn_wmma_f32_16x16x4_f32(false, a, false, b6, (short)0, acc[6], false, false);
            acc[7] = __builtin_amdgcn_wmma_f32_16x16x4_f32(false, a, false, b7, (short)0, acc[7], false, false);
        }
    }

    #pragma unroll
    for (int t = 0; t < 8; t++) {
        int cidx = wave * 128 + t * 16 + lm;
        if (cidx < NC) {
            #pragma unroll
            for (int v = 0; v < 8; v++) {
                int r = tile_m * 16 + v + 8 * half;
                int tg = tgtw[r];
                bool eq = row_eq[(size_t)tg * NC + cidx] != 0;
                float val = eq ? -INFINITY : acc[t][v] * rscale[r] * INV_TEMP;
                out[(size_t)r * OUTW + 1 + cidx] = val;
            }
        }
    }
}

extern "C" void kernel_launch(void* const* d_in, const int* in_sizes, int n_in,
                              void* d_out, int out_size, void* d_ws, size_t ws_size,
                              hipStream_t stream) {
    const float* x      = (const float*)d_in[0];   // [16,1500,768]
    const int*   label  = (const int*)d_in[1];     // [16,1500]
    const int*   mask_m = (const int*)d_in[2];     // [12000]
    const int*   mask_u = (const int*)d_in[3];     // [12000]
    const float* W      = (const float*)d_in[4];   // [768,256]
    const float* b      = (const float*)d_in[5];   // [256]
    const float* emb    = (const float*)d_in[6];   // [500,256]
    float* out = (float*)d_out;                    // [2*12000*501]

    char* ws = (char*)d_ws;
    size_t off = 0;
    float* pg    = (float*)(ws + off); off += (size_t)NROWS * DF * sizeof(float);     // 24.58 MB
    float* embn  = (float*)(ws + off); off += (size_t)NC * DF * sizeof(float);        // 512 KB
    float* Wpack = (float*)(ws + off); off += (size_t)KB1 * 4 * 32 * 8 * sizeof(float);  // 768 KB
    float* Epack = (float*)(ws + off); off += (size_t)KB2 * 4 * 32 * 16 * sizeof(float); // 512 KB
    float* rscl  = (float*)(ws + off); off += (size_t)NROWS * sizeof(float);          // 96 KB
    int*   tgtw  = (int*)(ws + off);   off += (size_t)NROWS * sizeof(int);            // 96 KB
    unsigned char* req = (unsigned char*)(ws + off);                                  // 250 KB

    emb_norm_kernel<<<NC, 32, 0, stream>>>(emb, embn);
    row_eq_kernel<<<(NC * NC + 255) / 256, 256, 0, stream>>>(emb, req);
    pack_w_kernel<<<(KB1 * 4 * 32 * 8) / 256, 256, 0, stream>>>(W, Wpack);
    pack_e_kernel<<<(KB2 * 4 * 32 * 16) / 256, 256, 0, stream>>>(embn, Epack);
    proj_gemm_kernel<<<NROWS / 16, 128, 0, stream>>>(x, mask_m, mask_u, Wpack, b, pg);
    norm_pos_kernel<<<NROWS / 8, 256, 0, stream>>>(pg, label, mask_m, mask_u, embn,
                                                   rscl, tgtw, out);
    logit_gemm_kernel<<<NROWS / 16, 128, 0, stream>>>(pg, Epack, rscl, tgtw, req, out);
}